// MLP_68066641707581
// MI455X (gfx1250) — compile-verified
//
#include <hip/hip_runtime.h>
#include <stdint.h>

// ---- problem constants (match reference) ----
#define FEAT_D    128
#define IN_DIM    384              // 3 * D
#define OUT_DIM   128
#define TILE_M    16               // rows per WMMA tile
#define SUPER_M   32               // edges per pipeline stage (2 WMMA tiles)
#define LDS_PITCH (IN_DIM + 4)     // 388 floats: 16B-aligned rows, bank-conflict-free
#define GRID_WGS  3750             // 18750 supertiles / 3750 = 5 each

typedef __attribute__((ext_vector_type(2)))  float v2f;
typedef __attribute__((ext_vector_type(8)))  float v8f;

// CDNA5 async global->LDS copy, 16 bytes. Tracked by ASYNCcnt.
// Operand order per ISA VGLOBAL fields: VDST = LDS byte address, VADDR = 64b global.
__device__ __forceinline__ void async_cp16(uint32_t ldsAddr, const float* g)
{
    asm volatile("global_load_async_to_lds_b128 %0, %1, off"
                 :: "v"(ldsAddr), "v"((uint64_t)(uintptr_t)g)
                 : "memory");
}

__device__ __forceinline__ void wait_async_keep12()
{   // wait until <=12 async ops outstanding (the 12 just issued for next stage)
    asm volatile("s_wait_asynccnt 0xc" ::: "memory");
}
__device__ __forceinline__ void wait_async_all()
{
    asm volatile("s_wait_asynccnt 0x0" ::: "memory");
}

// Stage one 32-edge supertile (rows = [e||u[src]||v[dst]], 384 f32) into LDS.
// 256 threads: 16 threads/row-group, 12 async B128 copies per thread.
__device__ __forceinline__ void stage_supertile(float* buf, int super,
                                                const float* __restrict__ e_feat,
                                                const float* __restrict__ u_feat,
                                                const float* __restrict__ v_feat,
                                                const int*   __restrict__ src_idx,
                                                const int*   __restrict__ dst_idx,
                                                int tid)
{
    const int r = tid >> 4;          // 0..15
    const int j = tid & 15;          // float4 slot
    #pragma unroll
    for (int g = 0; g < 2; ++g) {
        const int row = g * 16 + r;              // 0..31
        const int e   = super * SUPER_M + row;
        const int su  = src_idx[e];
        const int dv  = dst_idx[e];
        const float* eR = e_feat + (size_t)e  * FEAT_D;
        const float* uR = u_feat + (size_t)su * FEAT_D;
        const float* vR = v_feat + (size_t)dv * FEAT_D;
        // generic->LDS: low 32 bits of a flat shared pointer == DS byte address
        const uint32_t l0 = (uint32_t)(uintptr_t)(&buf[row * LDS_PITCH]);
        async_cp16(l0 + (uint32_t)(j +  0) * 16u, eR + (j +  0) * 4);
        async_cp16(l0 + (uint32_t)(j + 16) * 16u, eR + (j + 16) * 4);
        async_cp16(l0 + (uint32_t)(j + 32) * 16u, uR + (j +  0) * 4);
        async_cp16(l0 + (uint32_t)(j + 48) * 16u, uR + (j + 16) * 4);
        async_cp16(l0 + (uint32_t)(j + 64) * 16u, vR + (j +  0) * 4);
        async_cp16(l0 + (uint32_t)(j + 80) * 16u, vR + (j + 16) * 4);
    }
}

__global__ __launch_bounds__(256)
void gnn_edge_mlp_wmma_f32(const float* __restrict__ e_feat,
                           const float* __restrict__ u_feat,
                           const float* __restrict__ v_feat,
                           const int*   __restrict__ src_idx,
                           const int*   __restrict__ dst_idx,
                           const float* __restrict__ W,    // [IN_DIM][OUT_DIM]
                           const float* __restrict__ bias, // [OUT_DIM]
                           float*       __restrict__ out,  // [N_E][OUT_DIM]
                           int n_e)
{
    __shared__ float ldsA[2][SUPER_M * LDS_PITCH];   // 2 x 48.5 KB double buffer

    const int tid  = threadIdx.x;
    const int wave = tid >> 5;         // 0..7 -> output column tile
    const int lane = tid & 31;
    const int half = lane >> 4;        // 0: lanes 0-15, 1: lanes 16-31
    const int nidx = lane & 15;
    const int col  = wave * 16 + nidx;

    const int nsuper = n_e / SUPER_M;  // 18750 for the reference shapes
    const int stride = gridDim.x;

    const float bv = bias[col];
    const float* wBase = W + (size_t)(2 * half) * OUT_DIM + col;

    int s = blockIdx.x;
    if (s >= nsuper) return;

    // prologue: stage first supertile into buffer 0
    stage_supertile(&ldsA[0][0], s, e_feat, u_feat, v_feat, src_idx, dst_idx, tid);

    int buf = 0;
    while (true) {
        const int snext = s + stride;
        const bool have_next = (snext < nsuper);
        if (have_next) {   // prefetch next stage into the other buffer
            stage_supertile(&ldsA[buf ^ 1][0], snext,
                            e_feat, u_feat, v_feat, src_idx, dst_idx, tid);
            wait_async_keep12();    // current buffer's copies complete (in-order)
        } else {
            wait_async_all();
        }
        __syncthreads();            // all threads' copies for `buf` visible

        // ---- WMMA: two 16x16 output tiles per wave, shared W fragments ----
        v8f c0 = { bv, bv, bv, bv, bv, bv, bv, bv };
        v8f c1 = c0;
        const float* a0 = &ldsA[buf][(     nidx) * LDS_PITCH + 2 * half];
        const float* a1 = &ldsA[buf][(16 + nidx) * LDS_PITCH + 2 * half];

        #pragma unroll 8
        for (int k = 0; k < IN_DIM; k += 4) {
            v2f wf;                                   // B 4x16 fragment
            wf.x = wBase[(size_t)(k + 0) * OUT_DIM];
            wf.y = wBase[(size_t)(k + 1) * OUT_DIM];
            v2f af0, af1;                             // A 16x4 fragments
            af0.x = a0[k]; af0.y = a0[k + 1];
            af1.x = a1[k]; af1.y = a1[k + 1];
            c0 = __builtin_amdgcn_wmma_f32_16x16x4_f32(false, af0, false, wf,
                                                       (short)0, c0, false, false);
            c1 = __builtin_amdgcn_wmma_f32_16x16x4_f32(false, af1, false, wf,
                                                       (short)0, c1, false, false);
        }

        // ---- non-temporal stores: don't let the 307MB stream thrash L2 ----
        const size_t edge0 = (size_t)s * SUPER_M;
        #pragma unroll
        for (int i = 0; i < 8; ++i) {
            const int m = i + 8 * half;               // C/D layout row
            __builtin_nontemporal_store(c0[i], &out[(edge0 +      m) * OUT_DIM + col]);
            __builtin_nontemporal_store(c1[i], &out[(edge0 + 16 + m) * OUT_DIM + col]);
        }

        __syncthreads();            // done reading `buf` before it is re-staged
        if (!have_next) break;
        s = snext;
        buf ^= 1;
    }
}

extern "C" void kernel_launch(void* const* d_in, const int* in_sizes, int n_in,
                              void* d_out, int out_size, void* d_ws, size_t ws_size,
                              hipStream_t stream)
{
    const float* e_feat = (const float*)d_in[0];
    const float* u_feat = (const float*)d_in[1];
    const float* v_feat = (const float*)d_in[2];
    const int*   src    = (const int*)  d_in[3];
    const int*   dst    = (const int*)  d_in[4];
    const float* W      = (const float*)d_in[5];
    const float* b      = (const float*)d_in[6];
    float*       out    = (float*)d_out;

    const int n_e    = in_sizes[3];            // 600000
    const int nsuper = n_e / SUPER_M;          // 18750
    const int grid   = (nsuper < GRID_WGS) ? nsuper : GRID_WGS;

    gnn_edge_mlp_wmma_f32<<<grid, 256, 0, stream>>>(
        e_feat, u_feat, v_feat, src, dst, W, b, out, n_e);
}